// SpatialBlock_24696061952234
// MI455X (gfx1250) — compile-verified
//
#include <hip/hip_runtime.h>
#include <math.h>

#define NN 100000
#define EE 1600000
#define FD 128
#define HH 8
#define CC 16

typedef __attribute__((ext_vector_type(2))) float v2f;
typedef __attribute__((ext_vector_type(8))) float v8f;
typedef __attribute__((ext_vector_type(4))) int   v4i;

#if __has_builtin(__builtin_amdgcn_global_load_async_to_lds_b128)
#define HAVE_ASYNC_LDS 1
typedef __attribute__((address_space(1))) v4i gv4i;   // global int4
typedef __attribute__((address_space(3))) v4i lv4i;   // LDS int4
#endif

// ---------------------------------------------------------------------------
// float atomic max via sign-split integer atomics (init value must be -inf)
// ---------------------------------------------------------------------------
__device__ __forceinline__ void atomicMaxF(float* addr, float v) {
    if (v >= 0.0f) {
        atomicMax(reinterpret_cast<int*>(addr), __float_as_int(v));
    } else {
        atomicMin(reinterpret_cast<unsigned int*>(addr), __float_as_uint(v));
    }
}

// ---------------------------------------------------------------------------
// Tiny 128x128 transpose so B-fragment pairs (k0,k0+1) are contiguous (b64)
// ---------------------------------------------------------------------------
__global__ __launch_bounds__(256) void transpose_w(const float* __restrict__ W,
                                                   float* __restrict__ WT) {
    const int t = blockIdx.x * 256 + threadIdx.x;   // 16384 total
    const int k = t >> 7;
    const int n = t & 127;
    WT[n * FD + k] = W[t];
}

// ---------------------------------------------------------------------------
// Fused dual projection: outs = X@Ws + bs, outd = X@Wd + bd.
// Block = 256 threads = 8 waves; block computes 16 rows x 128 cols for BOTH
// weight matrices from a single LDS X tile. fp32 WMMA 16x16x4, K-loop of 32.
//
// fp32 WMMA VGPR layouts (wave32):
//  A (16x4):  lane<16: v0=A[l][k0+0], v1=A[l][k0+1];  lane>=16: k0+2, k0+3
//  B (4x16):  lane<16: v0=B[k0+0][n], v1=B[k0+1][n];  lane>=16: k0+2, k0+3
//  C/D (16x16): vgpr r, lane<16 -> D[r][lane]; lane>=16 -> D[r+8][lane-16]
// ---------------------------------------------------------------------------
__global__ __launch_bounds__(256) void gemm_proj2(const float* __restrict__ X,
                                                  const float* __restrict__ WTs,
                                                  const float* __restrict__ WTd,
                                                  const float* __restrict__ bs,
                                                  const float* __restrict__ bd,
                                                  float* __restrict__ outs,
                                                  float* __restrict__ outd) {
    __shared__ float xs[16][132];   // 132 % 64 = 4 -> conflict-free column reads

    const int tid     = threadIdx.x;
    const int wave    = tid >> 5;
    const int lane    = tid & 31;
    const int rowBase = blockIdx.x * 16;

#if defined(HAVE_ASYNC_LDS)
    {
#pragma unroll
        for (int i = 0; i < 2; ++i) {
            int q = tid + i * 256;    // 512 x 16B chunks = 16x128 tile
            int f = q << 2;
            int r = f >> 7;
            int c = f & 127;
            gv4i* g = (gv4i*)(X + (size_t)(rowBase + r) * FD + c);
            lv4i* l = (lv4i*)(&xs[r][c]);
            __builtin_amdgcn_global_load_async_to_lds_b128(g, l, 0, 0);
        }
#if __has_builtin(__builtin_amdgcn_s_wait_asynccnt)
        __builtin_amdgcn_s_wait_asynccnt(0);
#else
        asm volatile("s_wait_asynccnt 0" ::: "memory");
#endif
    }
#else
#pragma unroll
    for (int i = 0; i < 2; ++i) {
        int q = tid + i * 256;
        int f = q << 2;
        int r = f >> 7;
        int c = f & 127;
        const float4 v = *reinterpret_cast<const float4*>(X + (size_t)(rowBase + r) * FD + c);
        *reinterpret_cast<float4*>(&xs[r][c]) = v;
    }
#endif
    __syncthreads();

    const int mrow  = lane & 15;
    const int khalf = (lane >> 4) << 1;          // 0 (lanes 0-15) or 2 (lanes 16-31)
    const int ncol  = (wave << 4) + (lane & 15);
    const float* bsrow = WTs + (size_t)ncol * FD;   // transposed: [n][k]
    const float* bdrow = WTd + (size_t)ncol * FD;

    v8f accS = {};
    v8f accD = {};
#pragma unroll
    for (int kk = 0; kk < 32; ++kk) {
        const int k0 = (kk << 2) + khalf;
        v2f a;
        a.x = xs[mrow][k0];
        a.y = xs[mrow][k0 + 1];

        const float2 b2 = *reinterpret_cast<const float2*>(bsrow + k0);
        v2f b; b.x = b2.x; b.y = b2.y;
        accS = __builtin_amdgcn_wmma_f32_16x16x4_f32(
            false, a, false, b, (short)0, accS, false, false);

        const float2 d2 = *reinterpret_cast<const float2*>(bdrow + k0);
        v2f bb; bb.x = d2.x; bb.y = d2.y;
        accD = __builtin_amdgcn_wmma_f32_16x16x4_f32(
            false, a, false, bb, (short)0, accD, false, false);
    }

    const float bvs = bs[ncol];
    const float bvd = bd[ncol];
    const int mOff  = (lane >> 4) << 3;
#pragma unroll
    for (int r = 0; r < 8; ++r) {
        const size_t o = (size_t)(rowBase + mOff + r) * FD + ncol;
        outs[o] = accS[r] + bvs;
        outd[o] = accD[r] + bvd;
    }
}

// ---------------------------------------------------------------------------
// Init workspace accumulators (ws is poisoned before timing)
// ---------------------------------------------------------------------------
__global__ __launch_bounds__(256) void init_ws(float* __restrict__ m,
                                               float* __restrict__ denom,
                                               float* __restrict__ agg) {
    const int t = blockIdx.x * 256 + threadIdx.x;
    if (t < NN * HH) {
        m[t]     = -INFINITY;
        denom[t] = 0.0f;
    }
    if (t < NN * FD) agg[t] = 0.0f;
}

// ---------------------------------------------------------------------------
// Per-(edge,head) GATv2 logit + segment max
// ---------------------------------------------------------------------------
__global__ __launch_bounds__(256) void edge_logits(const float* __restrict__ xsrc,
                                                   const float* __restrict__ xdst,
                                                   const int*   __restrict__ ei,
                                                   const float* __restrict__ ea,
                                                   const float* __restrict__ We,
                                                   const float* __restrict__ att,
                                                   float* __restrict__ logits,
                                                   float* __restrict__ m) {
    const int t = blockIdx.x * 256 + threadIdx.x;   // < E*H = 12.8M
    const int h = t & 7;
    const int e = t >> 3;
    const int s = ei[e];
    const int d = ei[EE + e];
    const float eav = ea[e];
    const float* ps = xsrc + (size_t)s * FD + h * CC;
    const float* pd = xdst + (size_t)d * FD + h * CC;
    const float* pw = We + h * CC;
    const float* pa = att + h * CC;

    float acc = 0.0f;
#pragma unroll
    for (int c = 0; c < CC; ++c) {
        float v = ps[c] + pd[c] + eav * pw[c];
        v = (v > 0.0f) ? v : 0.2f * v;      // LeakyReLU(0.2)
        acc = fmaf(pa[c], v, acc);
    }
    logits[t] = acc;
    atomicMaxF(&m[d * HH + h], acc);
}

// ---------------------------------------------------------------------------
// Fused: ex = exp(logit - m[dst]); denom += ex; agg[dst] += ex * xsrc[src].
// Normalization by denom is linear per (node,head) and deferred to node_ln.
// 128 threads per edge (one per feature); only one lane per head adds denom.
// ---------------------------------------------------------------------------
__global__ __launch_bounds__(256) void edge_exp_agg(const float* __restrict__ xsrc,
                                                    const int*   __restrict__ ei,
                                                    const float* __restrict__ lg,
                                                    const float* __restrict__ m,
                                                    float* __restrict__ denom,
                                                    float* __restrict__ agg) {
    const int t = blockIdx.x * 256 + threadIdx.x;   // < E*128
    const int e = t >> 7;
    const int f = t & 127;
    const int h = f >> 4;
    const int s = ei[e];
    const int d = ei[EE + e];

    float mv = m[d * HH + h];
    if (!isfinite(mv)) mv = 0.0f;           // mirror reference fixup
    const float ex = expf(lg[e * HH + h] - mv);

    if ((f & 15) == 0) atomicAdd(&denom[d * HH + h], ex);
    atomicAdd(&agg[(size_t)d * FD + f], ex * xsrc[(size_t)s * FD + f]);
}

// ---------------------------------------------------------------------------
// Per-node: normalize by denom, + bias, LayerNorm, ELU, residual.
// One wave per node, 4 features per lane, wave32 shuffle reduction.
// ---------------------------------------------------------------------------
__global__ __launch_bounds__(256) void node_ln(const float* __restrict__ agg,
                                               const float* __restrict__ denom,
                                               const float* __restrict__ bias,
                                               const float* __restrict__ gamma,
                                               const float* __restrict__ beta,
                                               const float* __restrict__ x,
                                               float* __restrict__ out) {
    const int wave = threadIdx.x >> 5;
    const int lane = threadIdx.x & 31;
    const int node = blockIdx.x * 8 + wave;   // 12500 * 8 = 100000 exactly

    const int fBase = lane << 2;              // 4 consecutive feats, single head
    const int h     = lane >> 2;
    const size_t rowOff = (size_t)node * FD;
    const float inv = 1.0f / (denom[node * HH + h] + 1e-16f);

    float v[4];
    float sum = 0.0f, sumsq = 0.0f;
#pragma unroll
    for (int i = 0; i < 4; ++i) {
        const float a = agg[rowOff + fBase + i] * inv + bias[fBase + i];
        v[i] = a;
        sum += a;
        sumsq = fmaf(a, a, sumsq);
    }
#pragma unroll
    for (int off = 16; off >= 1; off >>= 1) {
        sum   += __shfl_xor(sum, off, 32);
        sumsq += __shfl_xor(sumsq, off, 32);
    }
    const float mu  = sum * (1.0f / FD);
    const float var = sumsq * (1.0f / FD) - mu * mu;
    const float rs  = rsqrtf(var + 1e-5f);

#pragma unroll
    for (int i = 0; i < 4; ++i) {
        const int f = fBase + i;
        float o = (v[i] - mu) * rs * gamma[f] + beta[f];
        o = (o > 0.0f) ? o : (expf(o) - 1.0f);   // ELU(alpha=1)
        out[rowOff + f] = o + x[rowOff + f];
    }
}

// ---------------------------------------------------------------------------
extern "C" void kernel_launch(void* const* d_in, const int* in_sizes, int n_in,
                              void* d_out, int out_size, void* d_ws, size_t ws_size,
                              hipStream_t stream) {
    const float* x      = (const float*)d_in[0];
    const int*   ei     = (const int*)  d_in[1];
    const float* ea     = (const float*)d_in[2];
    const float* W_src  = (const float*)d_in[3];
    const float* b_src  = (const float*)d_in[4];
    const float* W_dst  = (const float*)d_in[5];
    const float* b_dst  = (const float*)d_in[6];
    const float* W_edge = (const float*)d_in[7];
    const float* att    = (const float*)d_in[8];
    const float* bias   = (const float*)d_in[9];
    const float* gamma  = (const float*)d_in[10];
    const float* beta   = (const float*)d_in[11];
    float* out = (float*)d_out;

    float* ws  = (float*)d_ws;
    float* xs  = ws;                              // N*128
    float* xd  = xs + (size_t)NN * FD;            // N*128
    float* lg  = xd + (size_t)NN * FD;            // E*8
    float* m   = lg + (size_t)EE * HH;            // N*8
    float* dn  = m + (size_t)NN * HH;             // N*8
    float* agg = dn + (size_t)NN * HH;            // N*128
    float* wts = agg + (size_t)NN * FD;           // 128*128 (W_src^T)
    float* wtd = wts + FD * FD;                   // 128*128 (W_dst^T)

    transpose_w<<<64, 256, 0, stream>>>(W_src, wts);
    transpose_w<<<64, 256, 0, stream>>>(W_dst, wtd);
    init_ws<<<(NN * FD + 255) / 256, 256, 0, stream>>>(m, dn, agg);
    gemm_proj2<<<NN / 16, 256, 0, stream>>>(x, wts, wtd, b_src, b_dst, xs, xd);
    edge_logits<<<(EE * HH) / 256, 256, 0, stream>>>(xs, xd, ei, ea, W_edge, att, lg, m);
    edge_exp_agg<<<(int)(((size_t)EE * FD) / 256), 256, 0, stream>>>(xs, ei, lg, m, dn, agg);
    node_ln<<<NN / 8, 256, 0, stream>>>(agg, dn, bias, gamma, beta, x, out);
}